// NETWORK_33285996544619
// MI455X (gfx1250) — compile-verified
//
#include <hip/hip_runtime.h>

#define NBN 8192
#define ICHUNK 512

typedef __attribute__((ext_vector_type(2))) float v2f;
typedef __attribute__((ext_vector_type(4))) float v4f;
typedef __attribute__((ext_vector_type(8))) float v8f;

// ---------------- kernel 0: zero the partial-sum workspace (N floats) ----------------
__global__ void NET_zero_partial(float* __restrict__ p) {
  int i = blockIdx.x * blockDim.x + threadIdx.x;
  if (i < NBN) p[i] = 0.0f;
}

// ---------------- kernel 1: GEMV partials via V_WMMA_F32_16X16X4_F32 ----------------
// grid = (NBN/128, NBN/ICHUNK), block = 256 (8 waves; each wave owns 16 columns j)
// A[m,k] = mat[i0+k, j0+m]  (mat = w + alpha*hebb), B[k,n] = yin[i0+k] (bcast over n)
// D[m,n] = sum_k mat[i0+k, j0+m] * yin[i0+k]  -> identical across n.
__global__ __launch_bounds__(256) void NET_gemv_wmma(
    const float* __restrict__ w, const float* __restrict__ alpha,
    const float* __restrict__ hebb, const float* __restrict__ yin,
    float* __restrict__ partial) {
  const int lane = threadIdx.x & 31;
  const int wave = threadIdx.x >> 5;
  const int j0   = blockIdx.x * 128 + wave * 16;
  const int m    = lane & 15;       // column offset within tile (A's M index)
  const int half = lane >> 4;       // 0: K=0/1,  1: K=2/3  (per ISA A/B layout)
  const int iBase = blockIdx.y * ICHUNK;

  v8f c = {};
  for (int k0 = iBase; k0 < iBase + ICHUNK; k0 += 4) {
    const int kA0 = k0 + (half << 1);
    const long long off0 = (long long)kA0 * NBN + (j0 + m);
    const long long off1 = off0 + NBN;

    v2f a, b;
    // streaming, zero-reuse traffic -> non-temporal
    a.x = __builtin_nontemporal_load(&w[off0]) +
          __builtin_nontemporal_load(&alpha[off0]) * __builtin_nontemporal_load(&hebb[off0]);
    a.y = __builtin_nontemporal_load(&w[off1]) +
          __builtin_nontemporal_load(&alpha[off1]) * __builtin_nontemporal_load(&hebb[off1]);
    // yin is tiny (32 KB) and reused by every block -> regular cached loads
    b.x = yin[kA0];
    b.y = yin[kA0 + 1];

    // 8 args: (neg_a, A, neg_b, B, c_mod, C, reuse_a, reuse_b)
    c = __builtin_amdgcn_wmma_f32_16x16x4_f32(false, a, false, b, (short)0, c,
                                              false, false);
  }

  // D layout: VGPR r holds (M=r, N=lane) for lanes 0-15 and (M=r+8, N=lane-16)
  // for lanes 16-31; all N columns are identical, so lanes with (lane&15)==0
  // hold the 16 per-column partial sums.
  if (m == 0) {
    const int base = j0 + (half ? 8 : 0);
#pragma unroll
    for (int r = 0; r < 8; ++r) atomicAdd(&partial[base + r], c[r]);
  }
}

// ---------------- kernel 2: yout = tanh(partial + input) ----------------
__global__ void NET_finish_yout(const float* __restrict__ partial,
                                const float* __restrict__ input,
                                float* __restrict__ yout) {
  int j = blockIdx.x * blockDim.x + threadIdx.x;
  if (j < NBN) yout[j] = tanhf(partial[j] + input[j]);
}

// ---------------- kernel 3: hebb_new = (1-eta)*hebb + eta*outer(yin,yout) ----------------
// float4 NT streaming; grid-stride.
__global__ __launch_bounds__(256) void NET_hebb_update(
    const v4f* __restrict__ hebb, const float* __restrict__ yin,
    const v4f* __restrict__ yout, const float* __restrict__ etaP,
    v4f* __restrict__ out) {
  const float eta = etaP[0];
  const float om  = 1.0f - eta;
  const long long total  = (long long)NBN * NBN / 4;   // 16M float4s
  const long long stride = (long long)gridDim.x * blockDim.x;
  for (long long idx = (long long)blockIdx.x * blockDim.x + threadIdx.x;
       idx < total; idx += stride) {
    const int i  = (int)(idx >> 11);   // row: (idx*4)/8192
    const int j4 = (int)(idx & 2047);  // float4 index within row
    const float yi = eta * yin[i];
    v4f h  = __builtin_nontemporal_load(&hebb[idx]);
    v4f yo = yout[j4];                 // 32 KB, cached
    v4f r;
    r.x = om * h.x + yi * yo.x;
    r.y = om * h.y + yi * yo.y;
    r.z = om * h.z + yi * yo.z;
    r.w = om * h.w + yi * yo.w;
    __builtin_nontemporal_store(r, &out[idx]);
  }
}

extern "C" void kernel_launch(void* const* d_in, const int* in_sizes, int n_in,
                              void* d_out, int out_size, void* d_ws, size_t ws_size,
                              hipStream_t stream) {
  (void)in_sizes; (void)n_in; (void)out_size; (void)ws_size;
  // setup_inputs order: input, yin, hebb, w, alpha, eta (all fp32)
  const float* input = (const float*)d_in[0];
  const float* yin   = (const float*)d_in[1];
  const float* hebb  = (const float*)d_in[2];
  const float* w     = (const float*)d_in[3];
  const float* alpha = (const float*)d_in[4];
  const float* eta   = (const float*)d_in[5];

  float* yout     = (float*)d_out;            // first N floats
  float* hebb_out = (float*)d_out + NBN;      // next N*N floats
  float* partial  = (float*)d_ws;             // N floats of scratch

  NET_zero_partial<<<NBN / 256, 256, 0, stream>>>(partial);

  dim3 g1(NBN / 128, NBN / ICHUNK);           // 64 x 16 = 1024 blocks, 8192 waves
  NET_gemv_wmma<<<g1, 256, 0, stream>>>(w, alpha, hebb, yin, partial);

  NET_finish_yout<<<NBN / 256, 256, 0, stream>>>(partial, input, yout);

  NET_hebb_update<<<4096, 256, 0, stream>>>(
      (const v4f*)hebb, yin, (const v4f*)yout, eta, (v4f*)hebb_out);
}